// SoftMoELayer_65687229825319
// MI455X (gfx1250) — compile-verified
//
#include <hip/hip_runtime.h>
#include <math.h>

// Problem constants (match reference)
constexpr int B  = 8;
constexpr int T  = 4096;
constexpr int D  = 1024;
constexpr int F  = 4096;
constexpr int E  = 8;
constexpr int S  = 4;
constexpr int NS = E * S;   // 32 router slots
constexpr int KP = 8;       // split-K partitions for the low-parallelism GEMMs

typedef __attribute__((ext_vector_type(2))) float v2f;
typedef __attribute__((ext_vector_type(8))) float v8f;

__device__ __forceinline__ v8f wmma_f32(v2f a, v2f b, v8f c) {
    // D = A(16x4 f32) * B(4x16 f32) + C(16x16 f32)
    return __builtin_amdgcn_wmma_f32_16x16x4_f32(
        /*neg_a=*/false, a, /*neg_b=*/false, b,
        /*c_mod=*/(short)0, c, /*reuse_a=*/false, /*reuse_b=*/false);
}

// ---------------------------------------------------------------------------
// 1) Router logits: logits[b,t,n] = sum_d x[b,t,d] * slots[n,d]
//    One wave -> 16 tokens x 32 slots (two 16x16 WMMA accumulators).
// ---------------------------------------------------------------------------
__global__ __launch_bounds__(256) void k_logits(const float* __restrict__ x,
                                                const float* __restrict__ slots,
                                                float* __restrict__ logits) {
    int w    = threadIdx.x >> 5;
    int lane = threadIdx.x & 31;
    int tile = blockIdx.x * 8 + w;          // B*T/16 = 2048 tiles
    int b    = tile / (T / 16);
    int t0   = (tile % (T / 16)) * 16;
    int h = lane >> 4, m = lane & 15;

    v8f acc0 = {}, acc1 = {};
    const float* xr = x + ((size_t)b * T + t0 + m) * D + 2 * h;
    const float* s0 = slots + (size_t)m * D + 2 * h;
    const float* s1 = slots + (size_t)(m + 16) * D + 2 * h;

    for (int k = 0; k < D; k += 4) {
        v2f a  = *(const v2f*)(xr + k);
        v2f b0 = *(const v2f*)(s0 + k);
        v2f b1 = *(const v2f*)(s1 + k);
        acc0 = wmma_f32(a, b0, acc0);
        acc1 = wmma_f32(a, b1, acc1);
    }
    float* o = logits + ((size_t)b * T + t0) * NS;
#pragma unroll
    for (int v = 0; v < 8; v++) {
        int row = v + 8 * h;
        o[(size_t)row * NS + m]      = acc0[v];
        o[(size_t)row * NS + m + 16] = acc1[v];
    }
}

// ---------------------------------------------------------------------------
// 2) combine = softmax over the 32 slots per token
// ---------------------------------------------------------------------------
__global__ __launch_bounds__(256) void k_combine(const float* __restrict__ logits,
                                                 float* __restrict__ comb) {
    size_t idx = (size_t)blockIdx.x * 256 + threadIdx.x;   // token over B*T
    const float* l = logits + idx * NS;
    float v[NS];
    float mx = -3.4e38f;
#pragma unroll
    for (int i = 0; i < NS; i++) { v[i] = l[i]; mx = fmaxf(mx, v[i]); }
    float s = 0.f;
#pragma unroll
    for (int i = 0; i < NS; i++) { v[i] = __expf(v[i] - mx); s += v[i]; }
    float inv = 1.0f / s;
    float* c = comb + idx * NS;
#pragma unroll
    for (int i = 0; i < NS; i++) c[i] = v[i] * inv;
}

// ---------------------------------------------------------------------------
// 3) dispatch = softmax over T per (b, slot n); write transposed [b][n][t]
// ---------------------------------------------------------------------------
__global__ __launch_bounds__(256) void k_dispatch(const float* __restrict__ logits,
                                                  float* __restrict__ dispT) {
    __shared__ float red[256];
    int bn = blockIdx.x;        // 0 .. B*NS-1 = 255
    int b = bn >> 5, n = bn & 31;
    const float* col = logits + (size_t)b * T * NS + n;
    int tid = threadIdx.x;

    float mloc = -3.4e38f;
    for (int t = tid; t < T; t += 256) mloc = fmaxf(mloc, col[(size_t)t * NS]);
    red[tid] = mloc; __syncthreads();
    for (int off = 128; off > 0; off >>= 1) {
        if (tid < off) red[tid] = fmaxf(red[tid], red[tid + off]);
        __syncthreads();
    }
    float mx = red[0]; __syncthreads();

    float sloc = 0.f;
    for (int t = tid; t < T; t += 256) sloc += __expf(col[(size_t)t * NS] - mx);
    red[tid] = sloc; __syncthreads();
    for (int off = 128; off > 0; off >>= 1) {
        if (tid < off) red[tid] += red[tid + off];
        __syncthreads();
    }
    float inv = 1.0f / red[0];

    float* drow = dispT + ((size_t)b * NS + n) * T;
    for (int t = tid; t < T; t += 256)
        drow[t] = __expf(col[(size_t)t * NS] - mx) * inv;
}

// ---------------------------------------------------------------------------
// 4a) slot_in partials (split-K over T): 8x parallelism vs monolithic GEMM.
//     part[p][b][n][d] = sum_{t in chunk p} dispT[b,n,t] * x[b,t,d]
// ---------------------------------------------------------------------------
__global__ __launch_bounds__(256) void k_slotin_part(const float* __restrict__ dispT,
                                                     const float* __restrict__ x,
                                                     float* __restrict__ part) {
    constexpr int TCH = T / KP;   // 512 tokens per partition
    int w    = threadIdx.x >> 5;
    int lane = threadIdx.x & 31;
    int tile = blockIdx.x * 8 + w;     // KP*B*(D/16) = 4096 tiles
    int p   = tile >> 9;               // / (B*64)
    int rem = tile & 511;
    int b   = rem >> 6;
    int d0  = (rem & 63) * 16;
    int h = lane >> 4, m = lane & 15;
    int k0 = p * TCH;

    v8f acc0 = {}, acc1 = {};
    const float* a0p = dispT + ((size_t)b * NS + m) * T + k0 + 2 * h;
    const float* a1p = dispT + ((size_t)b * NS + m + 16) * T + k0 + 2 * h;
    const float* xp  = x + (size_t)b * T * D + (size_t)(k0 + 2 * h) * D + d0 + m;

    for (int k = 0; k < TCH; k += 4) {
        v2f a0 = *(const v2f*)(a0p + k);
        v2f a1 = *(const v2f*)(a1p + k);
        v2f bv;
        bv.x = xp[(size_t)k * D];
        bv.y = xp[(size_t)k * D + D];
        acc0 = wmma_f32(a0, bv, acc0);
        acc1 = wmma_f32(a1, bv, acc1);
    }
    float* o = part + (size_t)p * (B * NS * D);
#pragma unroll
    for (int v = 0; v < 8; v++) {
        int r = v + 8 * h;
        o[((size_t)b * NS + r) * D + d0 + m]      = acc0[v];
        o[((size_t)b * NS + r + 16) * D + d0 + m] = acc1[v];
    }
}

// 4b) deterministic fixed-order reduction of the split-K partials
__global__ __launch_bounds__(256) void k_red_slotin(const float* __restrict__ part,
                                                    float* __restrict__ si) {
    size_t i = (size_t)blockIdx.x * 256 + threadIdx.x;   // B*NS*D elements
    float s = 0.f;
#pragma unroll
    for (int p = 0; p < KP; p++) s += part[(size_t)p * (B * NS * D) + i];
    si[i] = s;
}

// ---------------------------------------------------------------------------
// 5) h = gelu(slot_in @ w1 + b1)   per expert: (32 x D) @ (D x F)
//    A rows r = b*4+s map to si[(b*32 + e*4 + s)*D + k]
// ---------------------------------------------------------------------------
__global__ __launch_bounds__(256) void k_ffn1(const float* __restrict__ si,
                                              const float* __restrict__ w1,
                                              const float* __restrict__ b1,
                                              float* __restrict__ hb) {
    int w    = threadIdx.x >> 5;
    int lane = threadIdx.x & 31;
    int tile = blockIdx.x * 8 + w;    // E*(F/16) = 2048 tiles
    int e  = tile >> 8;
    int n0 = (tile & 255) * 16;
    int h = lane >> 4, m = lane & 15;

    int r0 = m, r1 = m + 16;
    const float* a0p = si + (size_t)((r0 >> 2) * NS + e * S + (r0 & 3)) * D + 2 * h;
    const float* a1p = si + (size_t)((r1 >> 2) * NS + e * S + (r1 & 3)) * D + 2 * h;
    const float* wp  = w1 + (size_t)e * D * F + (size_t)(2 * h) * F + n0 + m;

    v8f acc0 = {}, acc1 = {};
    for (int k = 0; k < D; k += 4) {
        if ((k & 255) == 0 && k + 256 < D)
            __builtin_prefetch(wp + (size_t)(k + 256) * F, 0, 0);
        v2f a0 = *(const v2f*)(a0p + k);
        v2f a1 = *(const v2f*)(a1p + k);
        v2f bv;
        bv.x = wp[(size_t)k * F];
        bv.y = wp[(size_t)k * F + F];
        acc0 = wmma_f32(a0, bv, acc0);
        acc1 = wmma_f32(a1, bv, acc1);
    }
    float bias = b1[(size_t)e * F + n0 + m];
#pragma unroll
    for (int v = 0; v < 8; v++) {
        int r = v + 8 * h;
        float g0 = acc0[v] + bias;
        float g1 = acc1[v] + bias;
        g0 = 0.5f * g0 * (1.0f + erff(g0 * 0.70710678118f));
        g1 = 0.5f * g1 * (1.0f + erff(g1 * 0.70710678118f));
        hb[((size_t)e * NS + r) * F + n0 + m]      = g0;
        hb[((size_t)e * NS + r + 16) * F + n0 + m] = g1;
    }
}

// ---------------------------------------------------------------------------
// 6a) slot_out partials (split-K over F): part2[p][e][r][d]
// ---------------------------------------------------------------------------
__global__ __launch_bounds__(256) void k_ffn2_part(const float* __restrict__ hb,
                                                   const float* __restrict__ w2,
                                                   float* __restrict__ part2) {
    constexpr int FCH = F / KP;   // 512
    int w    = threadIdx.x >> 5;
    int lane = threadIdx.x & 31;
    int tile = blockIdx.x * 8 + w;    // KP*E*(D/16) = 4096 tiles
    int p   = tile >> 9;
    int rem = tile & 511;
    int e   = rem >> 6;
    int d0  = (rem & 63) * 16;
    int h = lane >> 4, m = lane & 15;
    int k0 = p * FCH;

    const float* a0p = hb + ((size_t)e * NS + m) * F + k0 + 2 * h;
    const float* a1p = hb + ((size_t)e * NS + m + 16) * F + k0 + 2 * h;
    const float* wp  = w2 + (size_t)e * F * D + (size_t)(k0 + 2 * h) * D + d0 + m;

    v8f acc0 = {}, acc1 = {};
    for (int k = 0; k < FCH; k += 4) {
        if ((k & 255) == 0 && k + 256 < FCH)
            __builtin_prefetch(wp + (size_t)(k + 256) * D, 0, 0);
        v2f a0 = *(const v2f*)(a0p + k);
        v2f a1 = *(const v2f*)(a1p + k);
        v2f bv;
        bv.x = wp[(size_t)k * D];
        bv.y = wp[(size_t)k * D + D];
        acc0 = wmma_f32(a0, bv, acc0);
        acc1 = wmma_f32(a1, bv, acc1);
    }
    float* o = part2 + (size_t)p * (E * NS * D);
#pragma unroll
    for (int v = 0; v < 8; v++) {
        int r = v + 8 * h;
        o[((size_t)e * NS + r) * D + d0 + m]      = acc0[v];
        o[((size_t)e * NS + r + 16) * D + d0 + m] = acc1[v];
    }
}

// 6b) reduce partials + bias b2 (fixed order -> deterministic)
__global__ __launch_bounds__(256) void k_red_ffn2(const float* __restrict__ part2,
                                                  const float* __restrict__ b2,
                                                  float* __restrict__ so) {
    size_t i = (size_t)blockIdx.x * 256 + threadIdx.x;   // E*NS*D elements
    float s = 0.f;
#pragma unroll
    for (int p = 0; p < KP; p++) s += part2[(size_t)p * (E * NS * D) + i];
    int e = (int)(i / (NS * D));
    int d = (int)(i % D);
    so[i] = s + b2[(size_t)e * D + d];
}

// ---------------------------------------------------------------------------
// 7) out[b,t,d] = sum_n comb[b,t,n] * slot_out[e(n), b*4+s(n), d]
//    K=32 -> A lives in registers; each wave sweeps a full 1024-wide D row.
// ---------------------------------------------------------------------------
__global__ __launch_bounds__(256) void k_out(const float* __restrict__ comb,
                                             const float* __restrict__ so,
                                             float* __restrict__ out) {
    int w    = threadIdx.x >> 5;
    int lane = threadIdx.x & 31;
    int tile = blockIdx.x * 8 + w;    // B*(T/16) = 2048 tiles
    int b  = tile >> 8;
    int t0 = (tile & 255) * 16;
    int h = lane >> 4, m = lane & 15;

    v2f a[8];
    const float* ap = comb + ((size_t)b * T + t0 + m) * NS + 2 * h;
#pragma unroll
    for (int kk = 0; kk < 8; kk++) a[kk] = *(const v2f*)(ap + 4 * kk);

    for (int d0 = 0; d0 < D; d0 += 16) {
        v8f acc = {};
#pragma unroll
        for (int kk = 0; kk < 8; kk++) {
            // n = 4*kk + 2h (+1): expert e = kk, slots s = 2h, 2h+1
            const float* bp = so + ((size_t)kk * NS + b * S + 2 * h) * D + d0 + m;
            v2f bv;
            bv.x = bp[0];
            bv.y = bp[D];
            acc = wmma_f32(a[kk], bv, acc);
        }
#pragma unroll
        for (int v = 0; v < 8; v++) {
            int row = t0 + v + 8 * h;
            out[((size_t)b * T + row) * D + d0 + m] = acc[v];
        }
    }
}

// ---------------------------------------------------------------------------
extern "C" void kernel_launch(void* const* d_in, const int* in_sizes, int n_in,
                              void* d_out, int out_size, void* d_ws, size_t ws_size,
                              hipStream_t stream) {
    const float* x     = (const float*)d_in[0];
    const float* slots = (const float*)d_in[1];
    const float* w1    = (const float*)d_in[2];
    const float* b1    = (const float*)d_in[3];
    const float* w2    = (const float*)d_in[4];
    const float* b2    = (const float*)d_in[5];
    float* out = (float*)d_out;

    float* ws      = (float*)d_ws;
    float* logits  = ws;                                  // B*T*NS        (4 MiB)
    float* comb    = logits  + (size_t)B * T * NS;        // B*T*NS        (4 MiB)
    float* dispT   = comb    + (size_t)B * T * NS;        // B*NS*T        (4 MiB)
    float* slot_in = dispT   + (size_t)B * NS * T;        // B*NS*D        (1 MiB)
    float* hbuf    = slot_in + (size_t)B * NS * D;        // E*NS*F        (4 MiB)
    float* slot_o  = hbuf    + (size_t)E * NS * F;        // E*NS*D        (1 MiB)
    float* part_si = slot_o  + (size_t)E * NS * D;        // KP*B*NS*D     (8 MiB)
    float* part_so = part_si + (size_t)KP * B * NS * D;   // KP*E*NS*D     (8 MiB)

    k_logits     <<<(B * T / 16) / 8,        256, 0, stream>>>(x, slots, logits);
    k_combine    <<<(B * T) / 256,           256, 0, stream>>>(logits, comb);
    k_dispatch   <<<B * NS,                  256, 0, stream>>>(logits, dispT);
    k_slotin_part<<<(KP * B * (D / 16)) / 8, 256, 0, stream>>>(dispT, x, part_si);
    k_red_slotin <<<(B * NS * D) / 256,      256, 0, stream>>>(part_si, slot_in);
    k_ffn1       <<<(E * (F / 16)) / 8,      256, 0, stream>>>(slot_in, w1, b1, hbuf);
    k_ffn2_part  <<<(KP * E * (D / 16)) / 8, 256, 0, stream>>>(hbuf, w2, part_so);
    k_red_ffn2   <<<(E * NS * D) / 256,      256, 0, stream>>>(part_so, b2, slot_o);
    k_out        <<<(B * (T / 16)) / 8,      256, 0, stream>>>(comb, slot_o, out);
}